// MutualAttention_395136991655
// MI455X (gfx1250) — compile-verified
//
#include <hip/hip_runtime.h>
#include <math.h>

// CDNA5 / gfx1250, wave32. f32 WMMA (16x16x4) for the row-dot GEMM so the
// softmax logits keep full f32 precision (they have magnitude ~1e3).

typedef __attribute__((ext_vector_type(2))) float v2f;
typedef __attribute__((ext_vector_type(8))) float v8f;

#define BB 32
#define NN 4096
#define DD 300
#define CC 100
#define CHUNKS 16
#define ROWS_PER_CHUNK (NN / CHUNKS) // 256

// Workspace layout (float offsets)
#define OFF_PART 0                                  // [2][BB][CHUNKS][DD]
#define OFF_S    (OFF_PART + 2 * BB * CHUNKS * DD)  // [2][BB][DD]
#define OFF_V    (OFF_S + 2 * BB * DD)              // [2][BB][DD*3]
#define OFF_SB   (OFF_V + 2 * BB * DD * 3)          // [2][BB]
#define OFF_A    (OFF_SB + 2 * BB)                  // [2][3][BB][NN]
// total = 1,170,496 floats (~4.7 MB)

// ---------------- Kernel 1a: partial column sums S[b,d] = sum_n x[b,n,d] ---
__global__ void colsum_stage1(const float* __restrict__ xu,
                              const float* __restrict__ xi,
                              float* __restrict__ ws) {
    int bid = blockIdx.x;                 // 2*BB*CHUNKS blocks
    int t   = bid / (BB * CHUNKS);
    int rem = bid % (BB * CHUNKS);
    int b   = rem / CHUNKS;
    int ch  = rem % CHUNKS;
    int d   = threadIdx.x;                // 320 threads, d < 300 active
    if (d >= DD) return;
    const float* x = (t == 0 ? xu : xi) + (size_t)b * NN * DD;
    const float* p = x + (size_t)(ch * ROWS_PER_CHUNK) * DD + d;
    float acc = 0.f;
#pragma unroll 4
    for (int r = 0; r < ROWS_PER_CHUNK; ++r) acc += p[(size_t)r * DD];
    ws[OFF_PART + (size_t)((t * BB + b) * CHUNKS + ch) * DD + d] = acc;
}

// ---------------- Kernel 1b: reduce chunk partials (fixed order) -----------
__global__ void colsum_stage2(float* __restrict__ ws) {
    int idx = blockIdx.x * blockDim.x + threadIdx.x; // over 2*BB*DD
    if (idx >= 2 * BB * DD) return;
    int t = idx / (BB * DD);
    int r = idx % (BB * DD);
    int b = r / DD;
    int d = r % DD;
    const float* p = ws + OFF_PART + (size_t)((t * BB + b) * CHUNKS) * DD + d;
    float s = 0.f;
    for (int ch = 0; ch < CHUNKS; ++ch) s += p[(size_t)ch * DD];
    ws[OFF_S + idx] = s;
}

// ---------------- Kernel 2: sum_u/sum_i -> v[b,d,k], scalar biases ---------
__global__ void build_v(const float* __restrict__ xu, const float* __restrict__ xi,
                        const float* __restrict__ wu, const float* __restrict__ bu,
                        const float* __restrict__ wi, const float* __restrict__ bi,
                        float* __restrict__ ws) {
    __shared__ float Su[DD], Si[DD], xu0[DD], xuL[DD], xi0[DD], xiL[DD];
    __shared__ float su[CC], si[CC];
    int b = blockIdx.x;
    int tid = threadIdx.x; // 128
    for (int d = tid; d < DD; d += blockDim.x) {
        Su[d]  = ws[OFF_S + (size_t)(0 * BB + b) * DD + d];
        Si[d]  = ws[OFF_S + (size_t)(1 * BB + b) * DD + d];
        xu0[d] = xu[(size_t)b * NN * DD + d];
        xuL[d] = xu[(size_t)b * NN * DD + (size_t)(NN - 1) * DD + d];
        xi0[d] = xi[(size_t)b * NN * DD + d];
        xiL[d] = xi[(size_t)b * NN * DD + (size_t)(NN - 1) * DD + d];
    }
    __syncthreads();
    if (tid < CC) {
        int c = tid;
        float au = 0.f, ai = 0.f;
        for (int d = 0; d < DD; ++d) {
            const float* wuc = wu + (size_t)(c * DD + d) * 3;
            const float* wic = wi + (size_t)(c * DD + d) * 3;
            // padded conv column-sum: k=0 misses last row, k=2 misses first
            au += (Su[d] - xuL[d]) * wuc[0] + Su[d] * wuc[1] + (Su[d] - xu0[d]) * wuc[2];
            ai += (Si[d] - xiL[d]) * wic[0] + Si[d] * wic[1] + (Si[d] - xi0[d]) * wic[2];
        }
        su[c] = au + (float)NN * bu[c];
        si[c] = ai + (float)NN * bi[c];
    }
    __syncthreads();
    // v_u[d,k] = sum_c sum_i[c]*w_u[c,d,k] ; v_i[d,k] = sum_c sum_u[c]*w_i[c,d,k]
    for (int idx = tid; idx < DD * 3; idx += blockDim.x) {
        float vu = 0.f, vi = 0.f;
        for (int c = 0; c < CC; ++c) {
            vu += si[c] * wu[(size_t)c * DD * 3 + idx];
            vi += su[c] * wi[(size_t)c * DD * 3 + idx];
        }
        ws[OFF_V + (size_t)(0 * BB + b) * DD * 3 + idx] = vu;
        ws[OFF_V + (size_t)(1 * BB + b) * DD * 3 + idx] = vi;
    }
    if (tid == 0) {
        float s0 = 0.f, s1 = 0.f;
        for (int c = 0; c < CC; ++c) { s0 += si[c] * bu[c]; s1 += su[c] * bi[c]; }
        ws[OFF_SB + 0 * BB + b] = s0;
        ws[OFF_SB + 1 * BB + b] = s1;
    }
}

// ---------------- Kernel 3: a_k[b,m] = x[16xD] * v[Dx3] via f32 WMMA -------
// B-operand kept in LDS padded to [D][16] with columns 3..15 zeroed, so every
// lane does an unconditional ds_load_2addr_b32 (no EXEC save/restore in the
// hot loop, EXEC stays all-ones for WMMA with zero per-iteration overhead).
__global__ void rowdot_wmma(const float* __restrict__ xu,
                            const float* __restrict__ xi,
                            float* __restrict__ ws) {
    __shared__ float vlds[DD * 16];  // 19.2 KB
    int bid = blockIdx.x;            // 2*BB*32 blocks
    int t   = bid / (BB * 32);
    int rem = bid % (BB * 32);
    int b   = rem / 32;
    int grp = rem % 32;
    int tid = threadIdx.x;           // 256 = 8 waves
    const float* vsrc = ws + OFF_V + (size_t)(t * BB + b) * DD * 3;
    for (int idx = tid; idx < DD * 16; idx += 256) {
        int d = idx >> 4;
        int n = idx & 15;
        vlds[idx] = (n < 3) ? vsrc[d * 3 + n] : 0.f;
    }
    __syncthreads();

    int wave = tid >> 5;
    int lane = tid & 31;
    int tile = grp * 8 + wave;       // 0..255, 16 rows each
    int rowBase = tile * 16;
    const float* x = (t == 0 ? xu : xi) + (size_t)b * NN * DD;

    // A layout (ISA 16x4 f32): lanes 0-15 hold (M=lane, K=0/1), lanes 16-31
    // hold (M=lane-16, K=2/3) -> each lane loads one aligned float2.
    int mrow = lane & 15;
    int hi   = lane >> 4;            // 0 or 1
    const float* arow = x + (size_t)(rowBase + mrow) * DD + (hi ? 2 : 0);
    int n = lane & 15;               // B/C/D column (filter tap for n<3)
    // B mirrors A: reg r, lanes 0-15 -> (K=r, N=lane); lanes 16-31 -> (K=2+r).
    const float* brow = vlds + (hi ? 2 * 16 : 0) + n;

    v8f acc = {};
    for (int d0 = 0; d0 < DD; d0 += 4) {
        v2f a = *reinterpret_cast<const v2f*>(arow + d0); // b64 load, 8B aligned
        v2f bf;
        bf.x = brow[(d0 + 0) * 16];
        bf.y = brow[(d0 + 1) * 16];
        acc = __builtin_amdgcn_wmma_f32_16x16x4_f32(
            /*neg_a=*/false, a, /*neg_b=*/false, bf,
            /*c_mod=*/(short)0, acc, /*reuse_a=*/false, /*reuse_b=*/false);
    }
    // C/D layout: VGPR j holds (M=j, N=lane) for lanes 0-15, (M=8+j, N=lane-16)
    // for lanes 16-31. Columns 0..2 are the three filter-tap row-dots.
    if (n < 3) {
        float* aout = ws + OFF_A + (size_t)((t * 3 + n) * BB + b) * NN
                      + rowBase + (hi ? 8 : 0);
#pragma unroll
        for (int j = 0; j < 8; ++j) aout[j] = acc[j];
    }
}

// ---------------- Kernel 4: logits = shift-combine, softmax over N ---------
__global__ void softmax_out(float* __restrict__ out, const float* __restrict__ ws) {
    __shared__ float l[NN];
    __shared__ float red[256];
    int bid = blockIdx.x;            // 2*BB
    int t = bid / BB;
    int b = bid % BB;
    int tid = threadIdx.x;           // 256
    float sbias = ws[OFF_SB + t * BB + b];
    const float* a0 = ws + OFF_A + (size_t)((t * 3 + 0) * BB + b) * NN;
    const float* a1 = ws + OFF_A + (size_t)((t * 3 + 1) * BB + b) * NN;
    const float* a2 = ws + OFF_A + (size_t)((t * 3 + 2) * BB + b) * NN;
    for (int nn = tid; nn < NN; nn += 256) {
        float v = a1[nn] + sbias;
        if (nn > 0) v += a0[nn - 1];
        if (nn < NN - 1) v += a2[nn + 1];
        l[nn] = v;
    }
    __syncthreads();
    float m = -3.402823466e38f;
    for (int nn = tid; nn < NN; nn += 256) m = fmaxf(m, l[nn]);
    red[tid] = m;
    __syncthreads();
    for (int s = 128; s > 0; s >>= 1) {
        if (tid < s) red[tid] = fmaxf(red[tid], red[tid + s]);
        __syncthreads();
    }
    m = red[0];
    __syncthreads();
    float se = 0.f;
    for (int nn = tid; nn < NN; nn += 256) se += expf(l[nn] - m);
    red[tid] = se;
    __syncthreads();
    for (int s = 128; s > 0; s >>= 1) {
        if (tid < s) red[tid] += red[tid + s];
        __syncthreads();
    }
    float inv = 1.f / red[0];
    float* o = out + (size_t)(t * BB + b) * NN;
    for (int nn = tid; nn < NN; nn += 256) o[nn] = expf(l[nn] - m) * inv;
}

extern "C" void kernel_launch(void* const* d_in, const int* in_sizes, int n_in,
                              void* d_out, int out_size, void* d_ws, size_t ws_size,
                              hipStream_t stream) {
    const float* xu = (const float*)d_in[0]; // [B, N, D]
    const float* xi = (const float*)d_in[1]; // [B, N, D]
    const float* wu = (const float*)d_in[2]; // [C, D, 3]
    const float* bu = (const float*)d_in[3]; // [C]
    const float* wi = (const float*)d_in[4]; // [C, D, 3]
    const float* bi = (const float*)d_in[5]; // [C]
    float* out = (float*)d_out;              // u_att [B,N] then i_att [B,N]
    float* ws  = (float*)d_ws;

    colsum_stage1<<<2 * BB * CHUNKS, 320, 0, stream>>>(xu, xi, ws);
    colsum_stage2<<<(2 * BB * DD + 255) / 256, 256, 0, stream>>>(ws);
    build_v<<<BB, 128, 0, stream>>>(xu, xi, wu, bu, wi, bi, ws);
    rowdot_wmma<<<2 * BB * 32, 256, 0, stream>>>(xu, xi, ws);
    softmax_out<<<2 * BB, 256, 0, stream>>>(out, ws);
}